// _IQ2XSQuantWeight_12945031430379
// MI455X (gfx1250) — compile-verified
//
#include <hip/hip_runtime.h>
#include <hip/hip_bf16.h>

typedef __attribute__((ext_vector_type(2))) float v2f;
typedef __attribute__((ext_vector_type(8))) float v8f;

#define NWAVES   8                 // waves per workgroup (256 threads = 8 wave32)
#define BPW      16                // blocks (of 8 elems) per wave == WMMA M tile
#define NGRID    512               // codebook entries
#define NTILES   (NGRID / 16)      // 32 codebook tiles of 16 codewords
#define NTHREADS (NWAVES * 32)

// Deterministic stand-in for the iq2xs grid table: 8-bit mask, bit i set ->
// q_i = 3, else q_i = 1.  ||q||^2 = 8 + 8*popcount(mask).
__device__ __forceinline__ unsigned iq2_grid_mask(unsigned c) {
    unsigned x = c * 2654435761u + 0x9E3779B9u;
    x ^= x >> 16; x *= 0x85EBCA6Bu; x ^= x >> 13;
    return x & 0xFFu;
}

__global__ __launch_bounds__(NTHREADS)
void iq2xs_quant_ste_kernel(const float* __restrict__ w,
                            float* __restrict__ out,
                            int nb)
{
    // Codebook tiles pre-staged in the exact f32 WMMA-B register image for the
    // two K=4 chunks.  Lane n (n<16) of tile t holds codeword c=t*16+n values
    // {q0,q1,q4,q5}*rsq ; lane n+16 holds {q2,q3,q6,q7}*rsq.  Dense: no zeros.
    __shared__ float4   ldsB[NTILES][32];    // 16 KB
    __shared__ unsigned bestLds[NWAVES][16];

    const int tid    = threadIdx.x;
    const int lane   = tid & 31;
    const int wave   = tid >> 5;
    const int lane16 = lane & 15;
    const int o      = (lane >> 4) * 2;      // this lane's element offset: 0 or 2

    // ---------------- build scaled codebook once per workgroup ------------------
    #pragma unroll
    for (int e = 0; e < (NTILES * 32) / NTHREADS; ++e) {
        const int idx = tid + e * NTHREADS;  // 0..1023
        const int t   = idx >> 5;
        const int ln  = idx & 31;
        const int oo  = (ln >> 4) * 2;
        const unsigned c   = (unsigned)(t * 16 + (ln & 15));
        const unsigned m   = iq2_grid_mask(c);
        const float    rsq = rsqrtf(8.0f + 8.0f * (float)__popc(m));
        const float    r3  = 3.0f * rsq;
        float4 d;
        d.x = ((m >> (oo + 0)) & 1u) ? r3 : rsq;   // K = oo
        d.y = ((m >> (oo + 1)) & 1u) ? r3 : rsq;   // K = oo+1
        d.z = ((m >> (oo + 4)) & 1u) ? r3 : rsq;   // K = oo+4
        d.w = ((m >> (oo + 5)) & 1u) ? r3 : rsq;   // K = oo+5
        ldsB[t][ln] = d;
    }
    __syncthreads();

    // ---------------- A tile: 16 blocks x K=8 magnitudes (f32) ------------------
    // V_WMMA_F32_16X16X4_F32 A layout: VGPR0 = K, VGPR1 = K+1; lanes 0-15 carry
    // K={0,1}, lanes 16-31 carry K={2,3}.  Two chunks cover K=0..7 exactly.
    const int mbase  = ((int)blockIdx.x * NWAVES + wave) * BPW;
    const int gblock = mbase + lane16;        // every lane maps to a block
    const bool valid = (gblock < nb);

    float wv[4];                              // this lane's 4 elements {o,o+1,o+4,o+5}
    #pragma unroll
    for (int i = 0; i < 4; ++i) wv[i] = 0.0f;

    if (valid) {
        const float* wb = w + (size_t)gblock * 8 + o;
        float2 p0 = *(const float2*)(wb);     // elems o, o+1
        float2 p1 = *(const float2*)(wb + 4); // elems o+4, o+5
        wv[0] = p0.x; wv[1] = p0.y; wv[2] = p1.x; wv[3] = p1.y;
    }
    v2f a0 = { fabsf(wv[0]), fabsf(wv[1]) };  // chunk0: K=0..3
    v2f a1 = { fabsf(wv[2]), fabsf(wv[3]) };  // chunk1: K=4..7

    // ---------------- codebook sweep: 2 f32 WMMAs per tile, B from LDS ----------
    unsigned best[8];
    #pragma unroll
    for (int r = 0; r < 8; ++r) best[r] = 0u;

    #pragma unroll 8
    for (int t = 0; t < NTILES; ++t) {
        const float4 d = ldsB[t][lane];       // ds_load_b128, immediate offset
        v2f b0 = { d.x, d.y };
        v2f b1 = { d.z, d.w };

        v8f cz = {};
        v8f c = __builtin_amdgcn_wmma_f32_16x16x4_f32(
                    false, a0, false, b0, (short)0, cz, false, false);
        c = __builtin_amdgcn_wmma_f32_16x16x4_f32(
                    false, a1, false, b1, (short)0, c,  false, false);

        const unsigned cidx = (unsigned)(t * 16) + (unsigned)lane16;
        #pragma unroll
        for (int r = 0; r < 8; ++r) {
            float s = c[r] * c[r];            // (dot/sqrt(norm))^2
            unsigned pk = (__float_as_uint(s) & 0xFFFFFE00u) | cidx;
            best[r] = (pk > best[r]) ? pk : best[r];
        }
    }

    // ---------------- cross-lane argmax within each 16-lane half ----------------
    #pragma unroll
    for (int r = 0; r < 8; ++r) {
        unsigned b = best[r];
        #pragma unroll
        for (int xm = 1; xm < 16; xm <<= 1) {
            unsigned q = __shfl_xor(b, xm, 32);
            b = (b > q) ? b : q;
        }
        best[r] = b;
    }

    // hand winners to owning lanes: lane0 has blocks 0..7, lane16 has blocks 8..15
    if (lane == 0) {
        #pragma unroll
        for (int r = 0; r < 8; ++r) bestLds[wave][r] = best[r];
    }
    if (lane == 16) {
        #pragma unroll
        for (int r = 0; r < 8; ++r) bestLds[wave][8 + r] = best[r];
    }
    __syncthreads();

    // ---------------- dequantize + store ----------------------------------------
    // Lane L and lane L+16 jointly own block mbase+lane16; each handles its 4 elems.
    if (valid) {
        const unsigned cbest = bestLds[wave][lane16] & (NGRID - 1u);
        const unsigned m     = iq2_grid_mask(cbest);
        const float    norm  = 8.0f + 8.0f * (float)__popc(m);

        float q[4];
        q[0] = ((m >> (o + 0)) & 1u) ? 3.0f : 1.0f;
        q[1] = ((m >> (o + 1)) & 1u) ? 3.0f : 1.0f;
        q[2] = ((m >> (o + 4)) & 1u) ? 3.0f : 1.0f;
        q[3] = ((m >> (o + 5)) & 1u) ? 3.0f : 1.0f;

        float pdot = fabsf(wv[0]) * q[0] + fabsf(wv[1]) * q[1]
                   + fabsf(wv[2]) * q[2] + fabsf(wv[3]) * q[3];
        // partner lane (xor 16) holds the other half of the block
        const float dot   = pdot + __shfl_xor(pdot, 16, 32);
        const float scale = dot / norm;

        float ov[4];
        #pragma unroll
        for (int i = 0; i < 4; ++i) {
            float sg = (float)((wv[i] > 0.0f) - (wv[i] < 0.0f));  // sign(0)=0
            ov[i] = sg * scale * q[i];
        }

        float* ob = out + (size_t)gblock * 8 + o;
        *(float2*)(ob)     = make_float2(ov[0], ov[1]);
        *(float2*)(ob + 4) = make_float2(ov[2], ov[3]);
    }
}

extern "C" void kernel_launch(void* const* d_in, const int* in_sizes, int n_in,
                              void* d_out, int out_size, void* d_ws, size_t ws_size,
                              hipStream_t stream) {
    const float* w   = (const float*)d_in[0];
    float*       out = (float*)d_out;
    const int n  = in_sizes[0];
    const int nb = n / 8;                      // 8-element blocks
    const int blocksPerWG = NWAVES * BPW;      // 128 blocks / workgroup
    const int ngrid = (nb + blocksPerWG - 1) / blocksPerWG;
    iq2xs_quant_ste_kernel<<<ngrid, NTHREADS, 0, stream>>>(w, out, nb);
}